// Spectral_Loss_2259152797805
// MI455X (gfx1250) — compile-verified
//
#include <hip/hip_runtime.h>
#include <hip/hip_bf16.h>
#include <math.h>

typedef __attribute__((ext_vector_type(2))) float v2f;
typedef __attribute__((ext_vector_type(8))) float v8f;

#define DFEAT 1024
#define WAVE 32

// ---------------------------------------------------------------------------
// Kernel 0: zero the three f64 accumulators [cross, diag_sum, diag_sq]
// ---------------------------------------------------------------------------
__global__ void sl_init_kernel(double* acc) {
    if (threadIdx.x < 3) acc[threadIdx.x] = 0.0;
}

// ---------------------------------------------------------------------------
// Kernel 1: per-row norms -> squared clip scales; per-row z1.z2 dot -> f64 acc
// One 256-thread block per row (row of both inputs).
// ---------------------------------------------------------------------------
__global__ __launch_bounds__(256) void sl_norm_kernel(
    const float* __restrict__ img, const float* __restrict__ txt,
    float* __restrict__ s1sq, float* __restrict__ s2sq, double* acc) {
    const int row = blockIdx.x;
    const float* r1 = img + (size_t)row * DFEAT;
    const float* r2 = txt + (size_t)row * DFEAT;

    float ss1 = 0.f, ss2 = 0.f, dp = 0.f;
    for (int i = threadIdx.x; i < DFEAT; i += 256) {
        float a = r1[i], b = r2[i];
        ss1 = fmaf(a, a, ss1);
        ss2 = fmaf(b, b, ss2);
        dp  = fmaf(a, b, dp);
    }
    // wave32 butterfly reduce
    for (int o = 16; o > 0; o >>= 1) {
        ss1 += __shfl_xor(ss1, o, WAVE);
        ss2 += __shfl_xor(ss2, o, WAVE);
        dp  += __shfl_xor(dp,  o, WAVE);
    }
    __shared__ float sm[3][8];
    const int w = threadIdx.x >> 5, l = threadIdx.x & 31;
    if (l == 0) { sm[0][w] = ss1; sm[1][w] = ss2; sm[2][w] = dp; }
    __syncthreads();
    if (threadIdx.x == 0) {
        float t1 = 0.f, t2 = 0.f, t3 = 0.f;
        for (int i = 0; i < 8; ++i) { t1 += sm[0][i]; t2 += sm[1][i]; t3 += sm[2][i]; }
        const float n1 = sqrtf(t1), n2 = sqrtf(t2);
        // clip: keep if ||z|| < sqrt(mu)=1 else scale to unit norm (mu = 1)
        const float sc1 = (n1 < 1.0f) ? 1.0f : 1.0f / fmaxf(n1, 1e-12f);
        const float sc2 = (n2 < 1.0f) ? 1.0f : 1.0f / fmaxf(n2, 1e-12f);
        s1sq[row] = sc1 * sc1;
        s2sq[row] = sc2 * sc2;
        const double dot = (double)sc1 * (double)sc2 * (double)t3;  // z1_i . z2_i
        atomicAdd(&acc[1], dot);          // diag sum   (for loss_part1)
        atomicAdd(&acc[2], dot * dot);    // diag sq sum (for off-diag removal)
    }
}

// ---------------------------------------------------------------------------
// WMMA f32 16x16x4 fragment helpers.
// A (16x4, MxK): lanes 0-15 hold K=0 (v0)/K=1 (v1); lanes 16-31 hold K=2/K=3.
// B (4x16, KxN): mirrored — lanes 0-15 rows K0/K1, lanes 16-31 rows K2/K3,
//                N striped across lanes within a VGPR.
// C/D (16x16):   v0..v7 = rows M=0..7 (lanes 0-15, N=lane) / M=8..15 (16-31).
// ---------------------------------------------------------------------------
__device__ __forceinline__ v8f wmma_f32(v2f a, v2f b, v8f c) {
    return __builtin_amdgcn_wmma_f32_16x16x4_f32(
        /*neg_a=*/false, a, /*neg_b=*/false, b,
        /*c_mod=*/(short)0, c, /*reuse_a=*/false, /*reuse_b=*/false);
}

// ---------------------------------------------------------------------------
// Kernel 2: cross = trace(G1*G2) = sum_ab (Z1^T Z1)_ab (Z2^T Z2)_ab.
// Each wave owns a 32x32 tile of the 1024x1024 Gram (2x2 WMMA tiles).
// Phase 1: accumulate G1 tile over K=N rows; phase 2: G2 tile; then
// elementwise product, wave reduce, one f64 atomic per wave.
// Clip scales are folded (squared) into the A operand only.
// ---------------------------------------------------------------------------
__global__ __launch_bounds__(256) void sl_gram_kernel(
    const float* __restrict__ raw1, const float* __restrict__ raw2,
    const float* __restrict__ s1sq, const float* __restrict__ s2sq,
    double* acc, int N) {
    const int wave  = blockIdx.x * (blockDim.x >> 5) + (threadIdx.x >> 5);
    const int lane  = threadIdx.x & 31;
    const int aBase = (wave >> 5) * 32;   // 32 tiles along a
    const int bBase = (wave & 31) * 32;   // 32 tiles along b
    const int halfk = lane >> 4;          // 0: K0/K1, 1: K2/K3
    const int cl    = lane & 15;          // column within 16-wide stripe

    v8f g1[2][2], g2[2][2];
#pragma unroll
    for (int i = 0; i < 2; ++i)
#pragma unroll
        for (int j = 0; j < 2; ++j) { g1[i][j] = (v8f)0.0f; g2[i][j] = (v8f)0.0f; }

    // ---- phase 1: G1 tile ----
#pragma unroll 4
    for (int k0 = 0; k0 < N; k0 += 4) {
        const int kr = k0 + 2 * halfk;
        const float* p0 = raw1 + (size_t)kr * DFEAT;
        const float* p1 = p0 + DFEAT;
        const float sA0 = s1sq[kr], sA1 = s1sq[kr + 1];
        v2f a0, a1, b0, b1;
        a0.x = p0[aBase + cl]       * sA0;  a0.y = p1[aBase + cl]       * sA1;
        a1.x = p0[aBase + 16 + cl]  * sA0;  a1.y = p1[aBase + 16 + cl]  * sA1;
        b0.x = p0[bBase + cl];              b0.y = p1[bBase + cl];
        b1.x = p0[bBase + 16 + cl];         b1.y = p1[bBase + 16 + cl];
        g1[0][0] = wmma_f32(a0, b0, g1[0][0]);
        g1[0][1] = wmma_f32(a0, b1, g1[0][1]);
        g1[1][0] = wmma_f32(a1, b0, g1[1][0]);
        g1[1][1] = wmma_f32(a1, b1, g1[1][1]);
    }

    // ---- phase 2: G2 tile ----
#pragma unroll 4
    for (int k0 = 0; k0 < N; k0 += 4) {
        const int kr = k0 + 2 * halfk;
        const float* p0 = raw2 + (size_t)kr * DFEAT;
        const float* p1 = p0 + DFEAT;
        const float sA0 = s2sq[kr], sA1 = s2sq[kr + 1];
        v2f a0, a1, b0, b1;
        a0.x = p0[aBase + cl]       * sA0;  a0.y = p1[aBase + cl]       * sA1;
        a1.x = p0[aBase + 16 + cl]  * sA0;  a1.y = p1[aBase + 16 + cl]  * sA1;
        b0.x = p0[bBase + cl];              b0.y = p1[bBase + cl];
        b1.x = p0[bBase + 16 + cl];         b1.y = p1[bBase + 16 + cl];
        g2[0][0] = wmma_f32(a0, b0, g2[0][0]);
        g2[0][1] = wmma_f32(a0, b1, g2[0][1]);
        g2[1][0] = wmma_f32(a1, b0, g2[1][0]);
        g2[1][1] = wmma_f32(a1, b1, g2[1][1]);
    }

    // ---- cross product of the two tiles ----
    double local = 0.0;
#pragma unroll
    for (int i = 0; i < 2; ++i)
#pragma unroll
        for (int j = 0; j < 2; ++j)
#pragma unroll
            for (int c = 0; c < 8; ++c)
                local += (double)g1[i][j][c] * (double)g2[i][j][c];

    for (int o = 16; o > 0; o >>= 1) local += __shfl_xor(local, o, WAVE);
    if (lane == 0) atomicAdd(&acc[0], local);
}

// ---------------------------------------------------------------------------
// Kernel 3: combine accumulators into the scalar loss
// ---------------------------------------------------------------------------
__global__ void sl_final_kernel(const double* acc, float* out, int N) {
    if (threadIdx.x == 0 && blockIdx.x == 0) {
        const double cross = acc[0], dsum = acc[1], dsq = acc[2];
        const double dN = (double)N;
        const double part1 = -2.0 * dsum / dN;
        const double part2 = (cross - dsq) / (dN * (dN - 1.0));
        out[0] = (float)(part1 + part2);  // mu = 1
    }
}

// ---------------------------------------------------------------------------
extern "C" void kernel_launch(void* const* d_in, const int* in_sizes, int n_in,
                              void* d_out, int out_size, void* d_ws, size_t ws_size,
                              hipStream_t stream) {
    const float* img = (const float*)d_in[0];
    const float* txt = (const float*)d_in[1];
    float* out = (float*)d_out;
    const int N = in_sizes[0] / DFEAT;   // 8192

    // workspace: [0,24)   three f64 accumulators (64B aligned region)
    //            [64, 64+4N)   s1sq   [64+4N, 64+8N) s2sq
    double* acc = (double*)d_ws;
    float* s1sq = (float*)((char*)d_ws + 64);
    float* s2sq = s1sq + N;

    sl_init_kernel<<<1, 32, 0, stream>>>(acc);
    sl_norm_kernel<<<N, 256, 0, stream>>>(img, txt, s1sq, s2sq, acc);
    // 1024 wave-tiles (32x32 grid of 32x32 tiles), 8 waves per 256-thread block
    sl_gram_kernel<<<128, 256, 0, stream>>>(img, txt, s1sq, s2sq, acc, N);
    sl_final_kernel<<<1, 32, 0, stream>>>(acc, out, N);
}